// LightForceRotary_61203283968197
// MI455X (gfx1250) — compile-verified
//
#include <hip/hip_runtime.h>

// ---------------------------------------------------------------------------
// LightForceRotary fused kernel for gfx1250 (MI455X)
//   offs[row] = tanh(x[row,:]·w + b) * 81.92       (GEMV via V_WMMA_F32_16X16X4_F32)
//   out = RoPE(x, pos = s + offs)                   (streamed from LDS tile)
// One workgroup (256 thr = 8 waves) processes 16 rows; x read once, out written once.
// ---------------------------------------------------------------------------

typedef __attribute__((ext_vector_type(2))) float v2f;
typedef __attribute__((ext_vector_type(8))) float v8f;

#define DIM        1024
#define SLEN       4096
#define ROWS_PB    16
#define PAD_STRIDE (DIM + 4)   // +4 floats: keeps rows 16B-aligned, conflict-free transposed reads
#define NTHREADS   256
#define NWAVES     8
#define KCHUNK     (DIM / NWAVES)   // 128 columns per wave
#define FORCE_SCALE (0.1f * 4096.0f * 0.2f)  // S*0.1*FORCE_FACTOR = 81.92

__global__ __launch_bounds__(NTHREADS)
void lightforce_rotary_kernel(const float* __restrict__ x,
                              const float* __restrict__ w,
                              const float* __restrict__ bmap,
                              const float* __restrict__ invf,
                              float* __restrict__ out)
{
    extern __shared__ float smem[];
    float* xt   = smem;                                   // [16][PAD_STRIDE] tile of x
    float* wl   = xt + ROWS_PB * PAD_STRIDE;              // [DIM] staged w
    float* part = wl + DIM;                               // [NWAVES][16] partial dots
    float* offs = part + NWAVES * ROWS_PB;                // [16] per-row offsets

    const int tid      = threadIdx.x;
    const int lane     = tid & 31;
    const int wave     = tid >> 5;
    const int row_base = blockIdx.x * ROWS_PB;

    // ---- Stage w (coalesced, one float4/thread) ----
    {
        const float4 wv = *(const float4*)(w + 4 * tid);
        *(float4*)(wl + 4 * tid) = wv;
    }
    // ---- Stage 16 rows of x into padded LDS tile (coalesced b128) ----
    #pragma unroll
    for (int r = 0; r < ROWS_PB; ++r) {
        const float4 v = *(const float4*)(x + (size_t)(row_base + r) * DIM + 4 * tid);
        *(float4*)(xt + r * PAD_STRIDE + 4 * tid) = v;
    }
    __syncthreads();

    // ---- Phase A: 16 dot products via f32 WMMA, waves split the K dimension ----
    // A(16x4) = w broadcast to every row:  lane half selects K pair {0,1} vs {2,3}.
    // B(4x16) = x tile transposed chunk:   lane supplies column N = lane&15.
    // Every row of D holds the same 16 partial dots -> acc[0], lanes 0..15.
    {
        const int   k0   = wave * KCHUNK;
        const int   koff = (lane < 16) ? 0 : 2;
        const float* xrow = xt + (lane & 15) * PAD_STRIDE;

        v8f acc = {};
        #pragma unroll
        for (int i = 0; i < KCHUNK / 4; ++i) {
            const int k = k0 + 4 * i + koff;
            v2f a, b;
            a.x = wl[k];
            a.y = wl[k + 1];
            b.x = xrow[k];
            b.y = xrow[k + 1];
            acc = __builtin_amdgcn_wmma_f32_16x16x4_f32(
                      /*neg_a=*/false, a, /*neg_b=*/false, b,
                      /*c_mod=*/(short)0, acc,
                      /*reuse_a=*/false, /*reuse_b=*/false);
        }
        if (lane < 16)
            part[wave * ROWS_PB + lane] = acc[0];
    }
    __syncthreads();

    // ---- Cross-wave reduce + tanh force ----
    if (tid < ROWS_PB) {
        float s = 0.0f;
        #pragma unroll
        for (int wv = 0; wv < NWAVES; ++wv)
            s += part[wv * ROWS_PB + tid];
        offs[tid] = tanhf(s + bmap[0]) * FORCE_SCALE;
    }
    __syncthreads();

    // ---- Phase B: rotary embedding, streamed from LDS, coalesced b128 stores ----
    // Thread t owns columns [4t, 4t+4) => pairs 2t and 2t+1, for all 16 rows.
    const float2 ifr   = *(const float2*)(invf + 2 * tid);  // inv_freq[2t], inv_freq[2t+1]
    const int    sbase = row_base & (SLEN - 1);             // s index of first row (S | row_base*16)

    #pragma unroll 4
    for (int r = 0; r < ROWS_PB; ++r) {
        const float pos = (float)(sbase + r) + offs[r];
        const float f0  = pos * ifr.x;
        const float f1  = pos * ifr.y;
        const float s0 = __sinf(f0), c0 = __cosf(f0);
        const float s1 = __sinf(f1), c1 = __cosf(f1);

        const float4 xv = *(const float4*)(xt + r * PAD_STRIDE + 4 * tid);
        float4 o;
        o.x = xv.x * c0 - xv.y * s0;
        o.y = xv.y * c0 + xv.x * s0;
        o.z = xv.z * c1 - xv.w * s1;
        o.w = xv.w * c1 + xv.z * s1;
        *(float4*)(out + (size_t)(row_base + r) * DIM + 4 * tid) = o;
    }
}

extern "C" void kernel_launch(void* const* d_in, const int* in_sizes, int n_in,
                              void* d_out, int out_size, void* d_ws, size_t ws_size,
                              hipStream_t stream) {
    const float* x    = (const float*)d_in[0];
    const float* w    = (const float*)d_in[1];
    const float* bmap = (const float*)d_in[2];
    const float* invf = (const float*)d_in[3];
    float*       out  = (float*)d_out;

    const int d    = in_sizes[1];              // 1024
    const int rows = in_sizes[0] / d;          // B*S = 16384

    const size_t shmem =
        (size_t)(ROWS_PB * PAD_STRIDE + DIM + NWAVES * ROWS_PB + ROWS_PB) * sizeof(float);

    // Dynamic LDS > 64KB: raise the runtime cap (host call, deterministic,
    // graph-capture safe — not a stream/alloc operation).
    (void)hipFuncSetAttribute((const void*)lightforce_rotary_kernel,
                              hipFuncAttributeMaxDynamicSharedMemorySize, (int)shmem);

    dim3 grid(rows / ROWS_PB);   // 1024 blocks, every block full -> EXEC all ones for WMMA
    dim3 block(NTHREADS);
    lightforce_rotary_kernel<<<grid, block, shmem, stream>>>(x, w, bmap, invf, out);
}